// EMDLoss_45475113730205
// MI455X (gfx1250) — compile-verified
//
#include <hip/hip_runtime.h>
#include <math.h>

// EMD greedy-matching loss for MI455X (gfx1250, wave32).
// pred/target: [8, 2048, 3] f32. Output: 1 f32 scalar.
//
// Design: 1 workgroup (8 waves, 256 threads) per batch. Each wave owns 16
// column-tiles (256 targets) whose B-fragments + squared norms live in VGPRs
// for the whole kernel. Per 16-row block, dot products are produced with
// V_WMMA_F32_16X16X4_F32 (K padded 3->4), converted to reduced cost
// c = |t|^2 - 2*dot (|p|^2 is row-constant, added at the end). The greedy
// argmin per row is a register scan + wave32 shfl_xor reduction + 8-wave LDS
// reduction. Used-target state is a per-lane 16-bit alive mask.

typedef __attribute__((ext_vector_type(2))) float v2f;
typedef __attribute__((ext_vector_type(8))) float v8f;

#define BATCH_N   8
#define NPTS      2048
#define WAVES_WG  8
#define TPB       256           // 8 waves * 32 lanes
#define TILES_PW  16            // tiles per wave: 16 tiles * 16 cols = 256 cols
#define ROWBLKS   (NPTS / 16)   // 128 row panels
#define BIGF      1e30f

__global__ __launch_bounds__(TPB)
void emd_greedy_wmma(const float* __restrict__ pred,
                     const float* __restrict__ tgt,
                     float* __restrict__ batch_sums) {
  const int b       = blockIdx.x;
  const int tid     = threadIdx.x;
  const int wave    = tid >> 5;
  const int lane    = tid & 31;
  const int laneLow = lane & 15;
  const bool hiHalf = lane >= 16;

  __shared__ float pnS[16];        // |p|^2 for the 16 rows of current panel
  __shared__ float wv[WAVES_WG];   // per-wave min value
  __shared__ int   wc[WAVES_WG];   // per-wave min column
  __shared__ int   bcastC;         // chosen column this step

  const float* __restrict__ predB = pred + (size_t)b * NPTS * 3;
  const float* __restrict__ tgtB  = tgt  + (size_t)b * NPTS * 3;

  // ---- Persistent B fragments (target^T tiles, K padded to 4) + col norms.
  // B 4x16 layout: VGPR0 = K0 (lanes 0-15) / K2 (lanes 16-31),
  //                VGPR1 = K1 (lanes 0-15) / K3==0 (lanes 16-31).
  const int colBase = wave * 256;
  v2f  bfrag[TILES_PW];
  float tn[TILES_PW];
#pragma unroll
  for (int t = 0; t < TILES_PW; ++t) {
    const int col = colBase + t * 16 + laneLow;
    const float tx = tgtB[col * 3 + 0];
    const float ty = tgtB[col * 3 + 1];
    const float tz = tgtB[col * 3 + 2];
    bfrag[t].x = hiHalf ? tz  : tx;
    bfrag[t].y = hiHalf ? 0.f : ty;
    tn[t] = tx * tx + ty * ty + tz * tz;
  }

  unsigned alive = 0xFFFFu;  // one bit per owned tile-column (this lane)
  float sum = 0.f;           // meaningful in tid 0 only

  for (int blk = 0; blk < ROWBLKS; ++blk) {
    const int rowBase = blk * 16;

    // ---- A fragment: 16 rows of pred, K padded to 4.
    const int row = rowBase + laneLow;
    const float px = predB[row * 3 + 0];
    const float py = predB[row * 3 + 1];
    const float pz = predB[row * 3 + 2];
    v2f afrag;
    afrag.x = hiHalf ? pz  : px;
    afrag.y = hiHalf ? 0.f : py;
    if (wave == 0 && !hiHalf) pnS[laneLow] = px * px + py * py + pz * pz;
    __syncthreads();  // pnS visible; previous panel fully consumed

    // ---- 16 WMMAs -> reduced costs c[t][r] = |t|^2 - 2*dot
    // D layout: row m -> VGPR (m&7), lanes 0-15 hold m<8, lanes 16-31 m>=8.
    float c[TILES_PW][8];
#pragma unroll
    for (int t = 0; t < TILES_PW; ++t) {
      v8f acc = {};
      acc = __builtin_amdgcn_wmma_f32_16x16x4_f32(
          /*neg_a=*/false, afrag, /*neg_b=*/false, bfrag[t],
          /*c_mod=*/(short)0, acc, /*reuse_a=*/false, /*reuse_b=*/false);
#pragma unroll
      for (int r = 0; r < 8; ++r)
        c[t][r] = __builtin_fmaf(-2.f, acc[r], tn[t]);
    }

    // ---- 16 sequential greedy steps over this panel
#pragma unroll
    for (int m = 0; m < 16; ++m) {
      const int  r      = m & 7;
      const bool active = (m < 8) ? !hiHalf : hiHalf;

      float v    = BIGF;
      int   cidx = 0x7FFFFFFF;
#pragma unroll
      for (int t = 0; t < TILES_PW; ++t) {
        const bool ok   = active && ((alive >> t) & 1u);
        const float cnd = c[t][r];
        const int  col  = colBase + t * 16 + laneLow;
        if (ok && cnd < v) { v = cnd; cidx = col; }
      }

      // wave32 min-with-index reduction (first-min tie-break on column)
#pragma unroll
      for (int s = 1; s < 32; s <<= 1) {
        const float ov = __shfl_xor(v, s, 32);
        const int   oc = __shfl_xor(cidx, s, 32);
        if (ov < v || (ov == v && oc < cidx)) { v = ov; cidx = oc; }
      }
      if (lane == 0) { wv[wave] = v; wc[wave] = cidx; }
      __syncthreads();

      if (tid == 0) {
        float bv = wv[0]; int bc = wc[0];
#pragma unroll
        for (int w = 1; w < WAVES_WG; ++w) {
          const float ov = wv[w]; const int oc = wc[w];
          if (ov < bv || (ov == bv && oc < bc)) { bv = ov; bc = oc; }
        }
        const float d2 = pnS[m] + bv;         // |p|^2 + |t|^2 - 2*dot
        sum += sqrtf(fmaxf(d2, 0.f));
        bcastC = bc;
      }
      __syncthreads();

      const int j = bcastC;                    // mark target j used
      if ((j >> 8) == wave && (j & 15) == laneLow)
        alive &= ~(1u << ((j >> 4) & 15));
    }
  }

  if (tid == 0) batch_sums[b] = sum;
}

__global__ void emd_finalize(const float* __restrict__ batch_sums,
                             float* __restrict__ out) {
  if (threadIdx.x == 0 && blockIdx.x == 0) {
    float s = 0.f;
#pragma unroll
    for (int b = 0; b < BATCH_N; ++b) s += batch_sums[b] / (float)NPTS;
    out[0] = s / (float)BATCH_N;
  }
}

extern "C" void kernel_launch(void* const* d_in, const int* in_sizes, int n_in,
                              void* d_out, int out_size, void* d_ws, size_t ws_size,
                              hipStream_t stream) {
  (void)in_sizes; (void)n_in; (void)out_size; (void)ws_size;
  const float* pred = (const float*)d_in[0];
  const float* tgt  = (const float*)d_in[1];
  float* sums = (float*)d_ws;      // BATCH_N floats of scratch
  float* out  = (float*)d_out;

  emd_greedy_wmma<<<BATCH_N, TPB, 0, stream>>>(pred, tgt, sums);
  emd_finalize<<<1, 32, 0, stream>>>(sums, out);
}